// LightGCNConv_69569880261266
// MI455X (gfx1250) — compile-verified
//
#include <hip/hip_runtime.h>
#include <stdint.h>

#define DIM 64
#define WAVE 32
#define WAVES_PER_BLOCK 8
#define BLOCK_THREADS (WAVE * WAVES_PER_BLOCK)
#define ROWS_PER_BUF 4                       // 4 edges per wave-iteration
#define BUF_FLOATS (ROWS_PER_BUF * DIM)      // 256 floats = 1 KB per buffer

// ---------------------------------------------------------------------------
// Kernel 1: out[0,:,:] = embeddings ; out[1..3,:,:] = 0   (float4 vectorized)
// ---------------------------------------------------------------------------
__global__ void lightgcn_init_kernel(const float* __restrict__ emb,
                                     float* __restrict__ out,
                                     long layer_elems /* N*64 */) {
    long i      = (long)blockIdx.x * blockDim.x + threadIdx.x;
    long stride = (long)gridDim.x * blockDim.x;
    long n4     = layer_elems >> 2;          // layer_elems divisible by 4
    long tot4   = n4 * 4;                    // 4 layers of float4s
    const float4* e4 = (const float4*)emb;
    float4* o4       = (float4*)out;
    const float4 z   = make_float4(0.f, 0.f, 0.f, 0.f);
    for (long k = i; k < tot4; k += stride) {
        o4[k] = (k < n4) ? e4[k] : z;
    }
}

// ---------------------------------------------------------------------------
// Kernel 2: COO SpMM  y[dst[e],:] += w[e] * x[src[e],:]
// One wave handles 4 edges per iteration. Gathered rows are staged in LDS via
// two async global->LDS b128 instructions (512 B each, double buffered on
// ASYNCcnt), then scattered with no-return f32 atomics into L2.
// ---------------------------------------------------------------------------
__global__ void lightgcn_spmm_kernel(const float* __restrict__ x,
                                     const float* __restrict__ ew,
                                     const int*   __restrict__ esrc,
                                     const int*   __restrict__ edst,
                                     float* __restrict__ y,
                                     int E) {
    // per wave: 2 buffers x 4 rows x 64 floats = 2 KB ; per block 16 KB
    __shared__ float lds[WAVES_PER_BLOCK * 2 * BUF_FLOATS];

    const int lane      = threadIdx.x & (WAVE - 1);
    const int waveInBlk = __builtin_amdgcn_readfirstlane((int)(threadIdx.x >> 5));
    const int waveId    = (int)blockIdx.x * WAVES_PER_BLOCK + waveInBlk;
    const int nWaves    = (int)gridDim.x * WAVES_PER_BLOCK;
    const int nQuads    = (E + 3) >> 2;

    if (waveId >= nQuads) return;

    float* wbase = &lds[waveInBlk * (2 * BUF_FLOATS)];
    // low 32 bits of a flat LDS pointer == byte offset within the WG's LDS
    const uint32_t wbase_off = (uint32_t)(uintptr_t)wbase;

    const int half  = lane >> 4;   // selects row within each 512 B async op
    const int qlane = lane & 15;   // 16 lanes x 16 B = one 256 B row

    // Two async b128 ops: op A stages rows(e0,e1), op B stages rows(e2,e3).
    auto prefetch = [&](int q, int buf) {
        int eb = 4 * q;
        int e0 = eb,     e1 = (eb + 1 < E) ? eb + 1 : E - 1;
        int e2 = (eb + 2 < E) ? eb + 2 : E - 1;
        int e3 = (eb + 3 < E) ? eb + 3 : E - 1;
        // wave-uniform adjacent indices -> wide scalar loads
        int s0 = esrc[e0], s1 = esrc[e1], s2 = esrc[e2], s3 = esrc[e3];
        int sA = half ? s1 : s0;
        int sB = half ? s3 : s2;
        uint64_t gA = (uint64_t)(uintptr_t)(x + (size_t)sA * DIM + (size_t)qlane * 4);
        uint64_t gB = (uint64_t)(uintptr_t)(x + (size_t)sB * DIM + (size_t)qlane * 4);
        uint32_t base = wbase_off + (uint32_t)buf * (BUF_FLOATS * 4u)
                      + (uint32_t)qlane * 16u;
        uint32_t lA = base + (uint32_t)half * (DIM * 4u);
        uint32_t lB = base + (uint32_t)(2 + half) * (DIM * 4u);
        asm volatile("global_load_async_to_lds_b128 %0, %1, off"
                     :: "v"(lA), "v"(gA) : "memory");
        asm volatile("global_load_async_to_lds_b128 %0, %1, off"
                     :: "v"(lB), "v"(gB) : "memory");
    };

    auto process = [&](int q, int buf) {
        const float* rows = wbase + buf * BUF_FLOATS;
        int eb = 4 * q;
#pragma unroll
        for (int j = 0; j < 4; ++j) {
            int e = eb + j;
            if (e < E) {
                float wj = ew[e];      // uniform adjacent -> merged s_loads
                int   dj = edst[e];
                float2 v = *(const float2*)(rows + j * DIM + lane * 2); // ds_load_b64
                float* o = y + (size_t)dj * DIM + (size_t)lane * 2;
                __hip_atomic_fetch_add(o,     wj * v.x, __ATOMIC_RELAXED,
                                       __HIP_MEMORY_SCOPE_AGENT);
                __hip_atomic_fetch_add(o + 1, wj * v.y, __ATOMIC_RELAXED,
                                       __HIP_MEMORY_SCOPE_AGENT);
            }
        }
    };

    int buf = 0;
    prefetch(waveId, 0);
    for (int q = waveId; q < nQuads; q += nWaves) {
        int qn = q + nWaves;
        if (qn < nQuads) {
            prefetch(qn, buf ^ 1);
            // 4 outstanding; oldest 2 (current buffer) must be complete
            asm volatile("s_wait_asynccnt 0x2" ::: "memory");
        } else {
            asm volatile("s_wait_asynccnt 0x0" ::: "memory");
        }
        process(q, buf);
        buf ^= 1;
    }
}

// ---------------------------------------------------------------------------
// Launch: init (copy+zero), then 3 chained SpMM layers, all on `stream`.
// Inputs: [0] embeddings (N*64 f32), [1] edge_weight (E f32),
//         [2] edge_src (E i32), [3] edge_dst (E i32)
// Output: [4, N, 64] f32 flat.
// ---------------------------------------------------------------------------
extern "C" void kernel_launch(void* const* d_in, const int* in_sizes, int n_in,
                              void* d_out, int out_size, void* d_ws, size_t ws_size,
                              hipStream_t stream) {
    const float* emb  = (const float*)d_in[0];
    const float* ew   = (const float*)d_in[1];
    const int*   esrc = (const int*)d_in[2];
    const int*   edst = (const int*)d_in[3];
    float* out = (float*)d_out;

    const long N     = (long)in_sizes[0] / DIM;
    const int  E     = in_sizes[1];
    const long layer = N * DIM;

    lightgcn_init_kernel<<<2048, BLOCK_THREADS, 0, stream>>>(emb, out, layer);

    int nQuads = (E + 3) >> 2;
    int blocks = (nQuads + WAVES_PER_BLOCK - 1) / WAVES_PER_BLOCK;
    if (blocks > 2048) blocks = 2048;
    if (blocks < 1) blocks = 1;

    for (int l = 1; l <= 3; ++l) {
        lightgcn_spmm_kernel<<<blocks, BLOCK_THREADS, 0, stream>>>(
            out + (l - 1) * layer, ew, esrc, edst, out + l * layer, E);
    }
}